// MeshTransformer_34059090657999
// MI455X (gfx1250) — compile-verified
//
#include <hip/hip_runtime.h>
#include <math.h>

#define B_  64
#define S_  32
#define P_  64
#define V_  2562
#define BS_ (B_*S_)        // 2048
#define VPB 128            // vertices per block in apply kernel (8 waves * 16)

typedef float v2f __attribute__((ext_vector_type(2)));
typedef float v8f __attribute__((ext_vector_type(8)));

// ---------------------------------------------------------------------------
// Kernel 1: per (b,s), reduce over P prototypes.
//   M[i][j]  = scale * sum_p w_p * R_p[i][j]   (R = Rx(a0)Ry(a1)Rz(a2))
//   tbar[k]  = sum_p w_p * t_p[k]
// Output: mt[bs*12 + {0..8}] = M row-major, mt[bs*12 + {9..11}] = tbar.
// ---------------------------------------------------------------------------
__global__ void mesh_reduce_kernel(const float* __restrict__ transforms,
                                   const float* __restrict__ weights,
                                   const float* __restrict__ scales,
                                   float* __restrict__ mt) {
    int bs = blockIdx.x * blockDim.x + threadIdx.x;
    if (bs >= BS_) return;

    float acc[12];
#pragma unroll
    for (int i = 0; i < 12; ++i) acc[i] = 0.f;

    const float* tr = transforms + (size_t)bs * (P_ * 6);
    const float* wp = weights    + (size_t)bs * P_;

    for (int p = 0; p < P_; ++p) {
        float tx = tr[p*6+0], ty = tr[p*6+1], tz = tr[p*6+2];
        float a0 = tr[p*6+3], a1 = tr[p*6+4], a2 = tr[p*6+5];
        float w  = wp[p];

        float sa, ca, sb, cb, sg, cg;
        __sincosf(a0, &sa, &ca);
        __sincosf(a1, &sb, &cb);
        __sincosf(a2, &sg, &cg);

        // R = Rx(a0) @ Ry(a1) @ Rz(a2), closed form
        float r00 =  cb*cg;             float r01 = -cb*sg;             float r02 =  sb;
        float r10 =  sa*sb*cg + ca*sg;  float r11 = -sa*sb*sg + ca*cg;  float r12 = -sa*cb;
        float r20 = -ca*sb*cg + sa*sg;  float r21 =  ca*sb*sg + sa*cg;  float r22 =  ca*cb;

        acc[0] = fmaf(w, r00, acc[0]); acc[1] = fmaf(w, r01, acc[1]); acc[2] = fmaf(w, r02, acc[2]);
        acc[3] = fmaf(w, r10, acc[3]); acc[4] = fmaf(w, r11, acc[4]); acc[5] = fmaf(w, r12, acc[5]);
        acc[6] = fmaf(w, r20, acc[6]); acc[7] = fmaf(w, r21, acc[7]); acc[8] = fmaf(w, r22, acc[8]);
        acc[9] = fmaf(w, tx, acc[9]);  acc[10] = fmaf(w, ty, acc[10]); acc[11] = fmaf(w, tz, acc[11]);
    }

    float sc = scales[bs];
    float* o = mt + (size_t)bs * 12;
#pragma unroll
    for (int i = 0; i < 9; ++i) o[i] = sc * acc[i];
    o[9] = acc[9]; o[10] = acc[10]; o[11] = acc[11];
}

// ---------------------------------------------------------------------------
// Kernel 2: out[bs, v, i] = sum_j verts[v,j] * M[i][j] + tbar[i]
// One (b,s) per blockIdx.y; 128 vertices per block; 8 waves, each wave does one
// V_WMMA_F32_16X16X4_F32 (A: 16 verts x [x,y,z,0], B: folded 3x3 in cols 0-2),
// then bounces D through LDS (adding tbar) so global stores are wide+contiguous.
// ---------------------------------------------------------------------------
__global__ void mesh_apply_kernel(const float* __restrict__ verts,
                                  const float* __restrict__ mt,
                                  float* __restrict__ out) {
    __shared__ __align__(16) float sV[VPB * 3];   // staged vertex chunk
    __shared__ __align__(16) float sM[12];        // M row-major + tbar
    __shared__ __align__(16) float sO[VPB * 3];   // transposed WMMA output

    const int tid = threadIdx.x;
    const int bs  = blockIdx.y;
    const int v0  = blockIdx.x * VPB;
    const bool full = (v0 + VPB) <= V_;           // uniform per block

    // ---- Stage vertex chunk (wide, coalesced) + the 12-float (b,s) record ----
    if (full) {
        if (tid < (VPB * 3) / 2) {                // 192 x b64 loads/stores
            const v2f* src = (const v2f*)(verts + (size_t)v0 * 3);
            ((v2f*)sV)[tid] = src[tid];
        }
    } else {
        const int nIn = (V_ - v0) * 3;
        for (int idx = tid; idx < VPB * 3; idx += blockDim.x)
            sV[idx] = (idx < nIn) ? verts[(size_t)v0 * 3 + idx] : 0.f;
    }
    if (tid < 12) sM[tid] = mt[(size_t)bs * 12 + tid];
    __syncthreads();

    const int lane = tid & 31;
    const int wave = tid >> 5;
    const int m    = lane & 15;     // A row / B column within the tile
    const int hi   = lane >> 4;     // 0 -> K={0,1}, 1 -> K={2,3}
    const int lv   = wave * 16 + m; // local vertex index (< VPB)

    // ---- A operand: unconditional LDS reads, value-select (no exec branches) ----
    // lanes 0-15: (x, y)   lanes 16-31: (z, 0)
    float ax0 = sV[lv * 3 + hi * 2];   // x or z
    float ay0 = sV[lv * 3 + 1];        // y (always safe)
    v2f a;
    a.x = ax0;
    a.y = (hi == 0) ? ay0 : 0.f;

    // ---- B operand: clamp index so loads are unconditional, then mask ----
    // col n<3: B[k][n] = M[n][k];  VGPR0 holds K=0 (lo lanes) / K=2 (hi lanes),
    //                              VGPR1 holds K=1 (lo lanes) / K=3=pad (hi lanes)
    const int mc = (m < 3) ? m : 0;    // clamped column
    float b0 = sM[mc * 3 + hi * 2];    // M[n][0] or M[n][2]
    float b1 = sM[mc * 3 + 1];         // M[n][1]
    const bool colOK = (m < 3);
    v2f b;
    b.x = colOK ? b0 : 0.f;
    b.y = (colOK && hi == 0) ? b1 : 0.f;

    v8f c = {};
    // 8 args: (neg_a, A, neg_b, B, c_mod, C, reuse_a, reuse_b)
    c = __builtin_amdgcn_wmma_f32_16x16x4_f32(false, a, false, b,
                                              (short)0, c, false, false);

    // ---- D -> LDS transpose (+tbar). D: VGPR r -> row r (lo lanes) / r+8 (hi) ----
    if (m < 3) {
        float tb = sM[9 + m];
#pragma unroll
        for (int r = 0; r < 8; ++r) {
            int row = r + hi * 8;  // vertex row within tile
            sO[(wave * 16 + row) * 3 + m] = c[r] + tb;
        }
    }
    __syncthreads();

    // ---- Wide, contiguous, non-temporal output stream ----
    // base (in floats) = bs*7686 + v0*3; only 8B-aligned in general -> b64 max.
    const size_t base = (size_t)bs * ((size_t)V_ * 3) + (size_t)v0 * 3;
    if (full) {
        if (tid < (VPB * 3) / 2) {                // 192 x GLOBAL_STORE_B64 (NT)
            v2f val = ((const v2f*)sO)[tid];
            __builtin_nontemporal_store(val, (v2f*)(out + base) + tid);
        }
    } else {
        const int nOut = (V_ - v0) * 3;
        for (int idx = tid; idx < nOut; idx += blockDim.x)
            __builtin_nontemporal_store(sO[idx], out + base + idx);
    }
}

// ---------------------------------------------------------------------------
extern "C" void kernel_launch(void* const* d_in, const int* in_sizes, int n_in,
                              void* d_out, int out_size, void* d_ws, size_t ws_size,
                              hipStream_t stream) {
    const float* verts      = (const float*)d_in[0];  // (V, 3)
    const float* scales     = (const float*)d_in[1];  // (B, S, 1)
    const float* transforms = (const float*)d_in[2];  // (B, S, P, 6)
    const float* weights    = (const float*)d_in[3];  // (B, S, P)
    float* out = (float*)d_out;                       // (B*S, V, 3)
    float* mt  = (float*)d_ws;                        // (BS, 12) = 96 KB scratch

    mesh_reduce_kernel<<<dim3((BS_ + 255) / 256), dim3(256), 0, stream>>>(
        transforms, weights, scales, mt);

    dim3 grid((V_ + VPB - 1) / VPB, BS_);
    mesh_apply_kernel<<<grid, dim3(256), 0, stream>>>(verts, mt, out);
}